// SpaceAttn_83554293776444
// MI455X (gfx1250) — compile-verified
//
#include <hip/hip_runtime.h>
#include <math.h>

typedef __attribute__((ext_vector_type(2))) float v2f;
typedef __attribute__((ext_vector_type(8))) float v8f;

#define B_    16
#define S_    2048
#define NST_  35
#define DD_   16
#define DF_   8
#define E_    16
#define H_    64
#define F_    88
#define N_    (S_*NST_)      // 71680 positions per batch
#define COLS_ (DD_*NST_)     // 560
#define MCH_  32             // mean s-chunks
#define SCH_  (S_/MCH_)      // 64
#define ACH_  70             // accumulation chunks (1024 positions each)

// workspace layout (float offsets)
#define WS_MEANP   0
#define WS_DMEAN   (WS_MEANP  + B_*MCH_*COLS_)     // B*560
#define WS_WVEC    (WS_DMEAN  + B_*COLS_)          // B*88
#define WS_CB      (WS_WVEC   + B_*F_)             // B
#define WS_DOTA    (WS_CB     + B_)                // B*35
#define WS_DOTW    (WS_DOTA   + B_*NST_)           // B*20
#define WS_MCONST  (WS_DOTW   + B_*20)             // B*35
#define WS_SCORES  (WS_MCONST + B_*NST_)           // B*N
#define WS_SMAX    (WS_SCORES + (size_t)B_*N_)     // B
#define WS_SSUM    (WS_SMAX   + B_)                // B
#define WS_PART    (WS_SSUM   + B_)                // B*ACH*130
#define WS_XBAR    (WS_PART   + B_*ACH_*130)       // B*88

// ---- K1: partial sums over s-chunks for dyn_mean -------------------------
__global__ void k_mean_partial(const float* __restrict__ dyn, float* __restrict__ ws) {
    int b = blockIdx.y, ch = blockIdx.x, col = threadIdx.x;
    if (col >= COLS_) return;
    const float* p = dyn + ((size_t)(b*S_ + ch*SCH_))*COLS_ + col;
    float s = 0.f;
    for (int i = 0; i < SCH_; ++i) s += p[(size_t)i*COLS_];
    ws[WS_MEANP + ((size_t)(b*MCH_ + ch))*COLS_ + col] = s;
}

// ---- K2: reduce chunks -> dyn_mean ---------------------------------------
__global__ void k_mean_reduce(float* __restrict__ ws) {
    int b = blockIdx.x, col = threadIdx.x;
    if (col >= COLS_) return;
    float s = 0.f;
    for (int ch = 0; ch < MCH_; ++ch)
        s += ws[WS_MEANP + ((size_t)(b*MCH_ + ch))*COLS_ + col];
    ws[WS_DMEAN + b*COLS_ + col] = s * (1.0f/(float)S_);
}

// ---- K3: per-batch folded weights: w_b, c_b, id dot-tables, mean consts --
__global__ void k_precompute(const float* __restrict__ hidden, const float* __restrict__ W,
                             const float* __restrict__ bA, const float* __restrict__ ea,
                             const float* __restrict__ ew, float* __restrict__ ws) {
    __shared__ float hs[H_], wv[F_];
    int b = blockIdx.x, tid = threadIdx.x;
    if (tid < H_) hs[tid] = hidden[b*H_ + tid] + hidden[B_*H_ + b*H_ + tid];  // sum over L=2
    __syncthreads();
    if (tid < F_) {
        float w = 0.f;
        for (int h = 0; h < H_; ++h) w += hs[h] * W[h*F_ + tid];
        wv[tid] = w;
        ws[WS_WVEC + b*F_ + tid] = w;
    }
    if (tid == 120) {
        float c = 0.f;
        for (int h = 0; h < H_; ++h) c += bA[h] * hs[h];
        ws[WS_CB + b] = c;
    }
    __syncthreads();
    if (tid < 35) {                       // aqst id dot table
        float v = 0.f;
        for (int e = 0; e < E_; ++e) v += ea[tid*E_ + e] * wv[56 + e];
        ws[WS_DOTA + b*35 + tid] = v;
    } else if (tid < 55) {                // weather id dot table
        int id = tid - 35; float v = 0.f;
        for (int e = 0; e < E_; ++e) v += ew[id*E_ + e] * wv[72 + e];
        ws[WS_DOTW + b*20 + id] = v;
    }
    if (tid >= 64 && tid < 99) {          // per-t mean constant: mean·(w_meanrep - w_dyntrans)
        int t = tid - 64; float v = 0.f;
        for (int d = 0; d < DD_; ++d)
            v += ws[WS_DMEAN + b*COLS_ + d*NST_ + t] * (wv[40 + d] - wv[8 + d]);
        ws[WS_MCONST + b*35 + t] = v;
    }
}

// ---- K4: score pass (GEVM per position, folded constants) ----------------
__global__ void k_scores(const float* __restrict__ fx, const float* __restrict__ dyn,
                         const float* __restrict__ dnan, const int* __restrict__ emb,
                         float* __restrict__ ws) {
    __shared__ float wv[F_], dA[35], dW[20], mc[35], cbv[1];
    int b = blockIdx.y, tid = threadIdx.x;
    if      (tid <  88) wv[tid]      = ws[WS_WVEC  + b*F_ + tid];
    else if (tid < 123) dA[tid-88]   = ws[WS_DOTA  + b*35 + (tid-88)];
    else if (tid < 143) dW[tid-123]  = ws[WS_DOTW  + b*20 + (tid-123)];
    else if (tid < 178) mc[tid-143]  = ws[WS_MCONST+ b*35 + (tid-143)];
    else if (tid == 178) cbv[0]      = ws[WS_CB + b];
    __syncthreads();
    int n = blockIdx.x*256 + tid;          // 280*256 == 71680 exactly
    int s = n / NST_, t = n - s*NST_;
    size_t base = (size_t)(b*S_ + s);
    int ida = emb[(base*2 + 0)*NST_ + t];
    int idw = emb[(base*2 + 1)*NST_ + t];
    float sc = cbv[0] + mc[t] + dA[ida] + dW[idw];
    const float* pf = fx  + base*(DF_*NST_) + t;
#pragma unroll
    for (int j = 0; j < DF_; ++j) sc += pf[j*NST_] * wv[j];
    const float* pd = dyn  + base*COLS_ + t;
    const float* pn = dnan + base*COLS_ + t;
#pragma unroll
    for (int j = 0; j < DD_; ++j) sc += pd[j*NST_] * wv[8 + j] + pn[j*NST_] * wv[24 + j];
    ws[WS_SCORES + (size_t)b*N_ + n] = sc;
}

// ---- K5: row max + sum(exp) per batch (scores are L2-resident) -----------
__global__ void k_softmax_stats(float* __restrict__ ws) {
    __shared__ float red[1024];
    int b = blockIdx.x, tid = threadIdx.x;
    const float* sc = ws + WS_SCORES + (size_t)b*N_;
    float m = -__builtin_inff();
    for (int i = tid; i < N_; i += 1024) m = fmaxf(m, sc[i]);
    red[tid] = m; __syncthreads();
    for (int o = 512; o > 0; o >>= 1) { if (tid < o) red[tid] = fmaxf(red[tid], red[tid+o]); __syncthreads(); }
    float mx = red[0]; __syncthreads();
    float s = 0.f;
    for (int i = tid; i < N_; i += 1024) s += __expf(sc[i] - mx);
    red[tid] = s; __syncthreads();
    for (int o = 512; o > 0; o >>= 1) { if (tid < o) red[tid] += red[tid+o]; __syncthreads(); }
    if (tid == 0) { ws[WS_SMAX + b] = mx; ws[WS_SSUM + b] = red[0]; }
}

// ---- K6: attn-weighted accumulation (regs for 40 fixed slots, LDS for ids)
__global__ void k_accum(const float* __restrict__ fx, const float* __restrict__ dyn,
                        const float* __restrict__ dnan, const int* __restrict__ emb,
                        float* __restrict__ ws) {
    __shared__ float acc[130];   // [0..7]=fix [8..23]=dyn [24..39]=nan [40..74]=acol [75..109]=aqstHist [110..129]=weatherHist
    int b = blockIdx.y, bx = blockIdx.x, tid = threadIdx.x;
    if (tid < 130) acc[tid] = 0.f;
    __syncthreads();
    float mx = ws[WS_SMAX + b];
    const float* sc = ws + WS_SCORES + (size_t)b*N_;
    float sf[8], sd[16], sn[16];
#pragma unroll
    for (int j = 0; j < 8; ++j)  sf[j] = 0.f;
#pragma unroll
    for (int j = 0; j < 16; ++j) { sd[j] = 0.f; sn[j] = 0.f; }
    for (int k = 0; k < 4; ++k) {
        int n = bx*1024 + k*256 + tid;     // 70*1024 == 71680
        float a = __expf(sc[n] - mx);
        int s = n / NST_, t = n - s*NST_;
        size_t base = (size_t)(b*S_ + s);
        int ida = emb[(base*2 + 0)*NST_ + t];
        int idw = emb[(base*2 + 1)*NST_ + t];
        atomicAdd(&acc[40 + t],  a);
        atomicAdd(&acc[75 + ida], a);
        atomicAdd(&acc[110 + idw], a);
        const float* pf = fx + base*(DF_*NST_) + t;
#pragma unroll
        for (int j = 0; j < 8; ++j) sf[j] += a * pf[j*NST_];
        const float* pd = dyn  + base*COLS_ + t;
        const float* pn = dnan + base*COLS_ + t;
#pragma unroll
        for (int j = 0; j < 16; ++j) { sd[j] += a * pd[j*NST_]; sn[j] += a * pn[j*NST_]; }
    }
#pragma unroll
    for (int j = 0; j < 8; ++j)  atomicAdd(&acc[j], sf[j]);
#pragma unroll
    for (int j = 0; j < 16; ++j) { atomicAdd(&acc[8 + j], sd[j]); atomicAdd(&acc[24 + j], sn[j]); }
    __syncthreads();
    if (tid < 130) ws[WS_PART + ((size_t)(b*ACH_ + bx))*130 + tid] = acc[tid];
}

// ---- K7: deterministic chunk reduce + assemble xbar[b][88] ---------------
__global__ void k_finalize(const float* __restrict__ ea, const float* __restrict__ ew,
                           float* __restrict__ ws) {
    __shared__ float tot[130];
    int b = blockIdx.x, tid = threadIdx.x;
    if (tid < 130) {
        float s = 0.f;
        for (int c = 0; c < ACH_; ++c) s += ws[WS_PART + ((size_t)(b*ACH_ + c))*130 + tid];
        tot[tid] = s;
    }
    __syncthreads();
    float inv = 1.0f / ws[WS_SSUM + b];
    float* Xb = ws + WS_XBAR + b*F_;
    if (tid < 8) Xb[tid] = tot[tid] * inv;
    else if (tid < 24) {
        int d = tid - 8; float sm = 0.f;
        for (int t = 0; t < NST_; ++t) sm += ws[WS_DMEAN + b*COLS_ + d*NST_ + t] * tot[40 + t];
        Xb[8 + d]  = (tot[8 + d] - sm) * inv;   // dyn_trans part
        Xb[40 + d] = sm * inv;                  // mean_rep part
    }
    else if (tid < 40) Xb[tid] = tot[tid] * inv;  // nan part
    else if (tid >= 56 && tid < 72) {
        int e = tid - 56; float v = 0.f;
        for (int id = 0; id < 35; ++id) v += tot[75 + id] * ea[id*E_ + e];
        Xb[tid] = v * inv;
    }
    else if (tid >= 72 && tid < 88) {
        int e = tid - 72; float v = 0.f;
        for (int id = 0; id < 20; ++id) v += tot[110 + id] * ew[id*E_ + e];
        Xb[tid] = v * inv;
    }
}

// ---- K8: Out(16x64) = Xbar(16x88) @ W^T(88x64) + bias via V_WMMA_F32_16X16X4_F32
__global__ void k_out_wmma(const float* __restrict__ W, const float* __restrict__ bA,
                           const float* __restrict__ ws, float* __restrict__ out) {
    const float* Xb = ws + WS_XBAR;
    int lane = threadIdx.x;
    int m  = lane & 15;            // A row / B col / D col index
    int kh = (lane >> 4) << 1;     // K sub-offset: lanes 0-15 -> K{0,1}, 16-31 -> K{2,3}
#pragma unroll
    for (int n0 = 0; n0 < H_; n0 += 16) {
        v8f c;
        float bias = bA[n0 + m];
#pragma unroll
        for (int i = 0; i < 8; ++i) c[i] = bias;   // C = bias broadcast over rows
#pragma unroll
        for (int k0 = 0; k0 < F_; k0 += 4) {
            v2f a, bb;
            a[0]  = Xb[m*F_ + k0 + kh];
            a[1]  = Xb[m*F_ + k0 + kh + 1];
            bb[0] = W[(n0 + m)*F_ + k0 + kh];      // B[k][n] = W_attn[n][k]
            bb[1] = W[(n0 + m)*F_ + k0 + kh + 1];
            c = __builtin_amdgcn_wmma_f32_16x16x4_f32(
                    false, a, false, bb, (short)0, c, false, false);
        }
#pragma unroll
        for (int i = 0; i < 8; ++i)
            out[(i + ((lane >> 4) << 3))*H_ + n0 + m] = c[i];  // (B,1,H) flat
    }
}

extern "C" void kernel_launch(void* const* d_in, const int* in_sizes, int n_in,
                              void* d_out, int out_size, void* d_ws, size_t ws_size,
                              hipStream_t stream) {
    const float* hidden = (const float*)d_in[0];
    const float* dyn    = (const float*)d_in[1];
    const float* dnan   = (const float*)d_in[2];
    const float* fx     = (const float*)d_in[3];
    const int*   emb    = (const int*)  d_in[4];
    const float* W      = (const float*)d_in[5];
    const float* bA     = (const float*)d_in[6];
    const float* ea     = (const float*)d_in[7];
    const float* ew     = (const float*)d_in[8];
    float* out = (float*)d_out;
    float* ws  = (float*)d_ws;

    k_mean_partial <<<dim3(MCH_, B_), 576, 0, stream>>>(dyn, ws);
    k_mean_reduce  <<<B_,             576, 0, stream>>>(ws);
    k_precompute   <<<B_,             128, 0, stream>>>(hidden, W, bA, ea, ew, ws);
    k_scores       <<<dim3(280, B_),  256, 0, stream>>>(fx, dyn, dnan, emb, ws);
    k_softmax_stats<<<B_,            1024, 0, stream>>>(ws);
    k_accum        <<<dim3(ACH_, B_), 256, 0, stream>>>(fx, dyn, dnan, emb, ws);
    k_finalize     <<<B_,             160, 0, stream>>>(ea, ew, ws);
    k_out_wmma     <<<1,               32, 0, stream>>>(W, bA, ws, out);
}